// VolumeAwareLoss_65704409694854
// MI455X (gfx1250) — compile-verified
//
#include <hip/hip_runtime.h>
#include <math.h>

// ---------------- problem constants ----------------
#define BB      2
#define CC      5
#define NN      96
#define VVOX    (NN * NN * NN)          // 884736
#define PLANE   (NN * NN)               // 9216
#define EPSF    1e-5f
#define SMOOTHF 1e-5f
#define BIGF    1e10f

// workspace layout (bytes)
#define G_BYTES   ((size_t)BB * 4 * VVOX * 4)   // 28,311,552  fp32 EDT array
#define M_BYTES   ((size_t)BB * VVOX)           //  1,769,472  packed pm/tm masks
// acc float slots
#define A_VOLS  0    // 10
#define A_INTER 10   // 8
#define A_SUMP  18   // 8
#define A_SUMG  26   // 8
#define A_FP    34   // 8
#define A_FN    42   // 8
#define A_CE    50   // 1
#define A_SURF  52   // 8
#define A_NPM   60   // 8
#define A_ANY   68   // 2 (uint bitfields)

typedef __attribute__((ext_vector_type(2))) float v2f;
typedef __attribute__((ext_vector_type(8))) float v8f;

// ---------------- wave32 f32 sum via V_WMMA_F32_16X16X4_F32 ----------------
// A[m][k]: lanes0-15 -> M=m, {VGPR0=K0, VGPR1=K1}; lanes16-31 -> {K2,K3}.
// With a=(v,0), B=ones: D[m][n] = v_m + v_{m+16}. D vgpr j holds row j (lanes0-15)
// and row j+8 (lanes16-31): sum(d0..d7) + xor-16 swap = full wave sum.
// EXEC must be all ones at the call site (all call sites keep full blocks active).
__device__ __forceinline__ float wave_sum(float v) {
#if __has_builtin(__builtin_amdgcn_wmma_f32_16x16x4_f32)
    v2f a; a[0] = v;   a[1] = 0.0f;
    v2f b; b[0] = 1.0f; b[1] = 1.0f;
    v8f c = {0.f, 0.f, 0.f, 0.f, 0.f, 0.f, 0.f, 0.f};
    v8f d = __builtin_amdgcn_wmma_f32_16x16x4_f32(false, a, false, b,
                                                  (short)0, c, false, false);
    float r = ((d[0] + d[1]) + (d[2] + d[3])) + ((d[4] + d[5]) + (d[6] + d[7]));
    // xor-16 swap via one ds_swizzle_b32 (group-of-32: and=0x1f, or=0, xor=0x10)
    r += __int_as_float(__builtin_amdgcn_ds_swizzle(__float_as_int(r), 0x401f));
    return r;
#else
    for (int off = 16; off; off >>= 1) v += __shfl_xor(v, off, 32);
    return v;
#endif
}

// ---------------- K0: zero accumulators ----------------
__global__ void k_zero(float* acc) {
    if (threadIdx.x < 128) acc[threadIdx.x] = 0.0f;
}

// ---------------- K1: fused softmax + reductions + mask emit ----------------
// grid (216, BB) x 256, 16 voxels/thread (216*256*16 = 884736)
__global__ __launch_bounds__(256) void k_main(const float* __restrict__ pred,
                                              const float* __restrict__ tgt,
                                              unsigned char* __restrict__ masks,
                                              float* __restrict__ acc) {
    __shared__ float    part[26][8];
    __shared__ unsigned apart[8];
    const int b     = blockIdx.y;
    const int chunk = blockIdx.x;
    const int tid   = threadIdx.x;
    const size_t pbase = (size_t)b * CC * VVOX;

    float lv[5]  = {0,0,0,0,0};
    float li[4]  = {0,0,0,0}, lsp[4] = {0,0,0,0}, lsg[4] = {0,0,0,0};
    float lfp[4] = {0,0,0,0}, lfn[4] = {0,0,0,0};
    float lce = 0.0f;
    unsigned ab = 0u;

    for (int it = 0; it < 16; ++it) {
        const int voxel = chunk * 4096 + it * 256 + tid;
        float x[5], t[5];
#pragma unroll
        for (int c = 0; c < 5; ++c) {
            x[c] = pred[pbase + (size_t)c * VVOX + voxel];
            t[c] = tgt [pbase + (size_t)c * VVOX + voxel];
        }
        float mx = x[0];
#pragma unroll
        for (int c = 1; c < 5; ++c) mx = fmaxf(mx, x[c]);
        float e[5], s = 0.0f;
#pragma unroll
        for (int c = 0; c < 5; ++c) { e[c] = expf(x[c] - mx); s += e[c]; }
        const float inv = 1.0f / s;
        const float ls  = logf(s);

        unsigned mb = 0u;
#pragma unroll
        for (int c = 0; c < 5; ++c) {
            lv[c] += t[c];
            lce   -= t[c] * (x[c] - mx - ls);   // -sum t*logp
        }
#pragma unroll
        for (int k = 0; k < 4; ++k) {
            const int c = k + 1;
            const float p = e[c] * inv;
            const float g = t[c];
            li[k]  += p * g;
            lsp[k] += p;
            lsg[k] += g;
            lfp[k] += p * (1.0f - g);
            lfn[k] += (1.0f - p) * g;
            if (p > 0.5f) mb |= (1u << k);
            if (g > 0.5f) mb |= (1u << (4 + k));
        }
        masks[(size_t)b * VVOX + voxel] = (unsigned char)mb;
        ab |= mb;
    }

    const int lane = tid & 31;
    const int wv   = tid >> 5;
    // wave-level wmma reductions -> per-wave partials in LDS
#pragma unroll
    for (int c = 0; c < 5; ++c) {
        float r = wave_sum(lv[c]);
        if (lane == 0) part[c][wv] = r;
    }
#pragma unroll
    for (int k = 0; k < 4; ++k) {
        float r0 = wave_sum(li[k]);
        float r1 = wave_sum(lsp[k]);
        float r2 = wave_sum(lsg[k]);
        float r3 = wave_sum(lfp[k]);
        float r4 = wave_sum(lfn[k]);
        if (lane == 0) {
            part[5  + k][wv] = r0;
            part[9  + k][wv] = r1;
            part[13 + k][wv] = r2;
            part[17 + k][wv] = r3;
            part[21 + k][wv] = r4;
        }
    }
    {
        float r = wave_sum(lce);
        if (lane == 0) part[25][wv] = r;
    }
    {
        int abi = (int)ab;
        for (int off = 16; off; off >>= 1) abi |= __shfl_xor(abi, off, 32);
        if (lane == 0) apart[wv] = (unsigned)abi;
    }
    __syncthreads();
    // cross-wave combine: one global atomic per quantity per block
    if (tid < 26) {
        float s = 0.0f;
#pragma unroll
        for (int w = 0; w < 8; ++w) s += part[tid][w];
        int slot;
        if      (tid < 5)  slot = A_VOLS  + b * 5 + tid;
        else if (tid < 9)  slot = A_INTER + b * 4 + (tid - 5);
        else if (tid < 13) slot = A_SUMP  + b * 4 + (tid - 9);
        else if (tid < 17) slot = A_SUMG  + b * 4 + (tid - 13);
        else if (tid < 21) slot = A_FP    + b * 4 + (tid - 17);
        else if (tid < 25) slot = A_FN    + b * 4 + (tid - 21);
        else               slot = A_CE;
        atomicAdd(&acc[slot], s);
    }
    if (tid == 0) {
        unsigned o = 0u;
#pragma unroll
        for (int w = 0; w < 8; ++w) o |= apart[w];
        atomicOr(((unsigned int*)acc) + A_ANY + b, o);
    }
}

// ---------------- K2: 6-neighbor erosion -> boundary masks + EDT init ----------------
// grid (3456, BB) x 256
__global__ __launch_bounds__(256) void k_bound(const unsigned char* __restrict__ masks,
                                               unsigned char* __restrict__ bmask,
                                               float* __restrict__ g) {
    const int b = blockIdx.y;
    const int voxel = blockIdx.x * 256 + threadIdx.x;
    const int d = voxel / PLANE;
    const int r = voxel % PLANE;
    const int h = r / NN;
    const int w = r % NN;
    const size_t base = (size_t)b * VVOX + voxel;

    const unsigned m = masks[base];
    unsigned e = m;
    e &= (d > 0)      ? (unsigned)masks[base - PLANE] : 0u;
    e &= (d < NN - 1) ? (unsigned)masks[base + PLANE] : 0u;
    e &= (h > 0)      ? (unsigned)masks[base - NN]    : 0u;
    e &= (h < NN - 1) ? (unsigned)masks[base + NN]    : 0u;
    e &= (w > 0)      ? (unsigned)masks[base - 1]     : 0u;
    e &= (w < NN - 1) ? (unsigned)masks[base + 1]     : 0u;
    const unsigned bd = m & ~e;
    bmask[base] = (unsigned char)bd;
#pragma unroll
    for (int k = 0; k < 4; ++k)
        g[((size_t)(b * 4 + k)) * VVOX + voxel] = ((bd >> (4 + k)) & 1u) ? 0.0f : BIGF;
}

// ---------------- K3a/b: tiled 1D squared-EDT along a strided axis (D or H) ----
// Lines run along `lineStride`; the stride-1 w dimension is tiled 32-wide so all
// global traffic is fully coalesced. Tile (96 x 32 fp32 = 12 KB) is staged with
// async global->LDS B128 (16 B per lane). Each thread produces 12 outputs of one
// w-column, reading one LDS value per j step.
// grid (3, 96, 8) x 256; blockIdx = (w-group, u, b*4+class)
__global__ __launch_bounds__(256) void k_edt_t(float* __restrict__ g,
                                               int lineStride, int uStride) {
    __shared__ float tile[NN][32];
    const int t  = threadIdx.x;
    const int wg = blockIdx.x;
    const int u  = blockIdx.y;
    const int bc = blockIdx.z;
    const size_t base = (size_t)bc * VVOX + (size_t)u * uStride + wg * 32;

#pragma unroll
    for (int k = 0; k < 3; ++k) {
        const int ch = t + k * 256;          // 768 chunks of 16B
        const int j  = ch >> 3;              // line position
        const int c4 = ch & 7;               // 4-float group within 32-wide row
        unsigned lds_off = (unsigned)(unsigned long long)(&tile[0][0])
                         + (unsigned)ch * 16u;
        unsigned long long ga =
            (unsigned long long)(g + base + (size_t)j * lineStride + c4 * 4);
        asm volatile("global_load_async_to_lds_b128 %0, %1, off"
                     :: "v"(lds_off), "v"(ga) : "memory");
    }
    asm volatile("s_wait_asynccnt 0" ::: "memory");
    __syncthreads();

    const int wcol  = t & 31;
    const int ibase = t >> 5;                // 0..7
    float m[12], fi[12];
#pragma unroll
    for (int k = 0; k < 12; ++k) {
        m[k]  = __builtin_inff();
        fi[k] = (float)(ibase + 8 * k);
    }
#pragma unroll 2
    for (int j = 0; j < NN; ++j) {
        const float v  = tile[j][wcol];
        const float fj = (float)j;
#pragma unroll
        for (int k = 0; k < 12; ++k) {
            const float dd = fi[k] - fj;
            m[k] = fminf(m[k], v + dd * dd);
        }
    }
#pragma unroll
    for (int k = 0; k < 12; ++k)
        g[base + (size_t)(ibase + 8 * k) * lineStride + wcol] = m[k];
}

// ---------------- K3c: 1D squared-EDT along W (stride-1 lines) ----------------
// grid 73728 x 96; per-lane async b32 staging is already coalesced.
__global__ __launch_bounds__(96) void k_edt_w(float* __restrict__ g) {
    __shared__ float line[NN];
    const int tid = threadIdx.x;
    const int L   = blockIdx.x;
    const size_t base = (size_t)(L / PLANE) * VVOX + (size_t)(L % PLANE) * NN;

    unsigned lds_off = (unsigned)(unsigned long long)(&line[tid]);
    unsigned long long ga = (unsigned long long)(g + base + tid);
    asm volatile("global_load_async_to_lds_b32 %0, %1, off"
                 :: "v"(lds_off), "v"(ga) : "memory");
    asm volatile("s_wait_asynccnt 0" ::: "memory");
    __syncthreads();

    const float fi = (float)tid;
    float m = __builtin_inff();
#pragma unroll 8
    for (int j = 0; j < NN; ++j) {
        const float dd = fi - (float)j;
        m = fminf(m, line[j] + dd * dd);
    }
    g[base + tid] = m;
}

// ---------------- K4: surface sums over boundary voxels ----------------
// grid (216, BB) x 256, 16 voxels/thread
__global__ __launch_bounds__(256) void k_surf(const unsigned char* __restrict__ bmask,
                                              const float* __restrict__ g,
                                              float* __restrict__ acc) {
    __shared__ float part[8][8];
    const int b     = blockIdx.y;
    const int chunk = blockIdx.x;
    const int tid   = threadIdx.x;
    float lsum[4] = {0,0,0,0};
    float lcnt[4] = {0,0,0,0};

    for (int it = 0; it < 16; ++it) {
        const int voxel = chunk * 4096 + it * 256 + tid;
        const unsigned bd = bmask[(size_t)b * VVOX + voxel];
#pragma unroll
        for (int k = 0; k < 4; ++k) {
            if ((bd >> k) & 1u) {
                lsum[k] += sqrtf(g[((size_t)(b * 4 + k)) * VVOX + voxel]);
                lcnt[k] += 1.0f;
            }
        }
    }
    const int lane = tid & 31;
    const int wv   = tid >> 5;
#pragma unroll
    for (int k = 0; k < 4; ++k) {
        float s = wave_sum(lsum[k]);
        float c = wave_sum(lcnt[k]);
        if (lane == 0) { part[k][wv] = s; part[4 + k][wv] = c; }
    }
    __syncthreads();
    if (tid < 8) {
        float s = 0.0f;
#pragma unroll
        for (int w = 0; w < 8; ++w) s += part[tid][w];
        const int slot = (tid < 4) ? (A_SURF + b * 4 + tid)
                                   : (A_NPM + b * 4 + (tid - 4));
        atomicAdd(&acc[slot], s);
    }
}

// ---------------- K5: finalize 16 outputs ----------------
__global__ void k_final(const float* __restrict__ acc, float* __restrict__ out) {
    if (threadIdx.x != 0) return;
    const float SW[5] = {1.0f, 2.0f, 1.5f, 2.5f, 1.5f};
    const float BL[5] = {0.0f, 2000.0f, 8000.0f, 1000.0f, 3000.0f};
    const float W4[4] = {2.0f, 1.5f, 2.5f, 1.5f};

    float eff[2][5];
    for (int b = 0; b < 2; ++b)
        for (int c = 0; c < 5; ++c) {
            const float v = acc[A_VOLS + b * 5 + c];
            const float mult = fminf(sqrtf(BL[c] / fmaxf(v, EPSF)), 2.5f);
            eff[b][c] = SW[c] * mult;
        }
    float dice_s = 0.0f, tl_s = 0.0f;
    for (int b = 0; b < 2; ++b)
        for (int k = 0; k < 4; ++k) {
            const float I  = acc[A_INTER + b * 4 + k];
            const float P  = acc[A_SUMP  + b * 4 + k];
            const float G  = acc[A_SUMG  + b * 4 + k];
            const float FP = acc[A_FP    + b * 4 + k];
            const float FN = acc[A_FN    + b * 4 + k];
            dice_s += 1.0f - (2.0f * I + SMOOTHF) / (P + G + SMOOTHF);
            tl_s   += 1.0f - (I + SMOOTHF) / (I + 0.3f * FP + 0.7f * FN + SMOOTHF);
        }
    const float dice = dice_s * 0.125f;
    const float tl   = tl_s * 0.125f;
    const float ce   = acc[A_CE] / (float)(BB * VVOX);
    const float dl   = 0.5f * dice + 0.5f * ce;

    float wsv[2];
    for (int b = 0; b < 2; ++b) {
        const unsigned anyb = ((const unsigned*)acc)[A_ANY + b];
        float s = 0.0f;
        for (int k = 0; k < 4; ++k) {
            const float num = acc[A_SURF + b * 4 + k];
            const float n   = acc[A_NPM  + b * 4 + k];
            float val = num / fmaxf(n, 1.0f);
            if (!isfinite(val)) val = 10.0f;
            const bool pa = (anyb >> k) & 1u;
            const bool ta = (anyb >> (4 + k)) & 1u;
            s += (pa && ta) ? W4[k] * val : ((pa != ta) ? W4[k] * 10.0f : 0.0f);
        }
        wsv[b] = s * 0.25f;
    }
    float tot[2];
    for (int b = 0; b < 2; ++b) tot[b] = 0.4f * dl + 0.4f * tl + 0.2f * wsv[b];

    out[0] = 0.5f * (tot[0] + tot[1]);
    out[1] = dl;
    out[2] = tl;
    out[3] = 0.5f * (wsv[0] + wsv[1]);
    for (int b = 0; b < 2; ++b)
        for (int c = 0; c < 5; ++c) {
            const float mx = fmaxf(eff[0][c], eff[1][c]);
            out[4 + b * 5 + c] = eff[b][c] / fmaxf(mx, EPSF);
        }
    out[14] = tot[0];
    out[15] = tot[1];
}

extern "C" void kernel_launch(void* const* d_in, const int* in_sizes, int n_in,
                              void* d_out, int out_size, void* d_ws, size_t ws_size,
                              hipStream_t stream) {
    const float* pred = (const float*)d_in[0];
    const float* tgt  = (const float*)d_in[1];
    float* out = (float*)d_out;

    char* ws = (char*)d_ws;
    float* g             = (float*)ws;
    unsigned char* masks = (unsigned char*)(ws + G_BYTES);
    unsigned char* bmask = masks + M_BYTES;
    float* acc           = (float*)(ws + G_BYTES + 2 * M_BYTES);

    k_zero <<<1, 128, 0, stream>>>(acc);
    k_main <<<dim3(216, BB), 256, 0, stream>>>(pred, tgt, masks, acc);
    k_bound<<<dim3(3456, BB), 256, 0, stream>>>(masks, bmask, g);
    // EDT axes in reference order: D, H, W
    k_edt_t<<<dim3(3, NN, BB * 4), 256, 0, stream>>>(g, PLANE, NN);   // along D
    k_edt_t<<<dim3(3, NN, BB * 4), 256, 0, stream>>>(g, NN, PLANE);   // along H
    k_edt_w<<<BB * 4 * PLANE, NN, 0, stream>>>(g);                    // along W
    k_surf <<<dim3(216, BB), 256, 0, stream>>>(bmask, g, acc);
    k_final<<<1, 32, 0, stream>>>(acc, out);
}